// CapsuleLayer_44195213476592
// MI455X (gfx1250) — compile-verified
//
#include <hip/hip_runtime.h>
#include <hip/hip_bf16.h>
#include <math.h>

typedef __attribute__((ext_vector_type(2))) float v2f;
typedef __attribute__((ext_vector_type(4))) float v4f;
typedef __attribute__((ext_vector_type(8))) float v8f;

// Problem sizes (fixed by reference)
constexpr int B_  = 256;   // batch
constexpr int IU  = 8;     // in_unit (i)
constexpr int C_  = 1152;  // in_channel (c)
constexpr int U_  = 10;    // num_unit (u)
constexpr int S_  = 16;    // unit_size (s)
constexpr int K_  = C_ * IU;          // 9216 = contraction axis (c,i)
constexpr int KSPLIT = 16;
constexpr int CCHUNK = (C_ / KSPLIT); // 72 channels per k-split chunk

// ---------------------------------------------------------------------------
// Kernel 1: pack x[b,i,c] -> Xt[k=(c*8+i), b]  and zero-init b_ij.
// ---------------------------------------------------------------------------
__global__ void pack_x_kernel(const float* __restrict__ x,
                              float* __restrict__ Xt,
                              float* __restrict__ bij) {
  int tid = blockIdx.x * blockDim.x + threadIdx.x;  // 0 .. K_*B_-1
  if (tid >= K_ * B_) return;
  int b = tid & (B_ - 1);
  int k = tid >> 8;          // c*8 + i
  int c = k >> 3;
  int i = k & 7;
  Xt[tid] = x[(b * IU + i) * C_ + c];
  if (tid < C_ * U_) bij[tid] = 0.0f;
}

// ---------------------------------------------------------------------------
// Kernel 2: c_ij[c,:] = softmax over u of b_ij[c,:]   (1152 x 10, tiny)
// ---------------------------------------------------------------------------
__global__ void softmax_u_kernel(const float* __restrict__ bij,
                                 float* __restrict__ cij) {
  int c = blockIdx.x * blockDim.x + threadIdx.x;
  if (c >= C_) return;
  float m = -3.4e38f;
  float e[U_];
  for (int u = 0; u < U_; ++u) m = fmaxf(m, bij[c * U_ + u]);
  float sum = 0.0f;
  for (int u = 0; u < U_; ++u) { e[u] = expf(bij[c * U_ + u] - m); sum += e[u]; }
  float inv = 1.0f / sum;
  for (int u = 0; u < U_; ++u) cij[c * U_ + u] = e[u] * inv;
}

// ---------------------------------------------------------------------------
// Kernel 3: s_j partial GEMM (per u): for each K=4 step,
//   A[s,k=(c,i)] = c_ij[c,u] * W[c,u,s,i]  (scale folded into A load, loaded ONCE)
//   B[k,b]       = Xt[k,b]                 (two adjacent 16-wide b tiles)
// Two accumulators per wave -> two independent WMMA chains, A traffic halved.
// wave32 f32 operand layout: lanes0-15 hold K0/K1 in VGPR0/1, lanes16-31 K2/K3.
// ---------------------------------------------------------------------------
__global__ __launch_bounds__(32) void sj_gemm_kernel(
    const float* __restrict__ W, const float* __restrict__ cij,
    const float* __restrict__ Xt, float* __restrict__ P) {
  const int npair = blockIdx.x;        // 0..7  -> b tiles 2*npair, 2*npair+1
  const int ks    = blockIdx.y;        // 0..KSPLIT-1
  const int u     = blockIdx.z;        // 0..9
  const int lane  = threadIdx.x;
  const int half  = lane >> 4;         // 0: K0/K1, 1: K2/K3
  const int l16   = lane & 15;
  const int b0    = npair * 32 + l16;  // tile-0 column
  const int b1    = b0 + 16;           // tile-1 column

  v8f acc0 = {}, acc1 = {};
  const int cbeg = ks * CCHUNK;
  for (int c = cbeg; c < cbeg + CCHUNK; ++c) {
    const float scale = cij[c * U_ + u];                     // SGPR broadcast
    const float* wrow = W + ((c * U_ + u) * S_ + l16) * IU;  // 8 i-values, 8B aligned
#pragma unroll
    for (int j = 0; j < 2; ++j) {      // two K=4 steps cover i=0..7 of channel c
      const int k0 = c * 8 + 4 * j + 2 * half;
      v2f a = *(const v2f*)(wrow + 4 * j + 2 * half);
      a *= scale;
      const float* xr = Xt + (size_t)k0 * B_;
      v2f bb0, bb1;
      bb0.x = xr[b0];      bb0.y = xr[B_ + b0];
      bb1.x = xr[b1];      bb1.y = xr[B_ + b1];
      acc0 = __builtin_amdgcn_wmma_f32_16x16x4_f32(
          false, a, false, bb0, (short)0, acc0, false, false);
      acc1 = __builtin_amdgcn_wmma_f32_16x16x4_f32(
          false, a, false, bb1, (short)0, acc1, false, false);
    }
  }
  // D layout: VGPR r -> (s = r + 8*half, b = col): 8 contiguous s per lane
  float* dst0 = P + (((ks * U_ + u) * B_ + b0) * S_) + 8 * half;
  *(v4f*)(dst0)     = *((const v4f*)&acc0);
  *(v4f*)(dst0 + 4) = *(((const v4f*)&acc0) + 1);
  float* dst1 = P + (((ks * U_ + u) * B_ + b1) * S_) + 8 * half;
  *(v4f*)(dst1)     = *((const v4f*)&acc1);
  *(v4f*)(dst1 + 4) = *(((const v4f*)&acc1) + 1);
}

// ---------------------------------------------------------------------------
// Kernel 4: reduce KSPLIT partials -> s_j[b,u,s], apply squash -> vout[b,u,s]
// Deterministic fixed-order reduction. vout = workspace v, or d_out on last iter.
// ---------------------------------------------------------------------------
__global__ void squash_kernel(const float* __restrict__ P,
                              float* __restrict__ vout) {
  int tid = blockIdx.x * blockDim.x + threadIdx.x;   // u*256 + b
  if (tid >= B_ * U_) return;
  int b = tid & (B_ - 1);
  int u = tid >> 8;
  v4f s0 = {}, s1 = {}, s2 = {}, s3 = {};
  for (int ks = 0; ks < KSPLIT; ++ks) {
    const float* row = P + ((ks * U_ + u) * B_ + b) * S_;
    s0 += *(const v4f*)(row + 0);
    s1 += *(const v4f*)(row + 4);
    s2 += *(const v4f*)(row + 8);
    s3 += *(const v4f*)(row + 12);
  }
  float s[S_];
  *(v4f*)(s + 0) = s0; *(v4f*)(s + 4) = s1;
  *(v4f*)(s + 8) = s2; *(v4f*)(s + 12) = s3;
  float sq = 0.0f;
#pragma unroll
  for (int j = 0; j < S_; ++j) sq += s[j] * s[j];
  const float coef = (sq / (1.0f + sq)) / sqrtf(sq + 1e-8f);
  float* dst = vout + (b * U_ + u) * S_;
#pragma unroll
  for (int j = 0; j < S_; ++j) dst[j] = s[j] * coef;
}

// ---------------------------------------------------------------------------
// Kernel 5: agreement GEMM (per u): G[u][m=(c,i)][s] = sum_b Xt[m,b] * v[b,u,s]
//   M = 9216 (2 tiles per wave, B operand shared), N = 16 (s), K = 256 (b)
// ---------------------------------------------------------------------------
__global__ __launch_bounds__(32) void g_gemm_kernel(
    const float* __restrict__ Xt, const float* __restrict__ v,
    float* __restrict__ G) {
  const int mpair = blockIdx.x;   // 0..287 -> m tiles 2*mpair, 2*mpair+1
  const int u     = blockIdx.y;   // 0..9
  const int lane  = threadIdx.x;
  const int half  = lane >> 4;
  const int l16   = lane & 15;
  const int m0    = mpair * 32 + l16;
  const int m1    = m0 + 16;

  v8f acc0 = {}, acc1 = {};
  for (int kb = 0; kb < B_; kb += 4) {
    const int kk = kb + 2 * half;
    v2f a0 = *(const v2f*)(Xt + (size_t)m0 * B_ + kk);   // A[m0, kk..kk+1]
    v2f a1 = *(const v2f*)(Xt + (size_t)m1 * B_ + kk);   // A[m1, kk..kk+1]
    v2f bb;                                              // B[k=b, n=s] (shared)
    bb.x = v[((kk + 0) * U_ + u) * S_ + l16];
    bb.y = v[((kk + 1) * U_ + u) * S_ + l16];
    acc0 = __builtin_amdgcn_wmma_f32_16x16x4_f32(
        false, a0, false, bb, (short)0, acc0, false, false);
    acc1 = __builtin_amdgcn_wmma_f32_16x16x4_f32(
        false, a1, false, bb, (short)0, acc1, false, false);
  }
  // D: M = r + 8*half (row within m-tile), N = l16 = s
  const float* a0p = (const float*)&acc0;
  const float* a1p = (const float*)&acc1;
#pragma unroll
  for (int r = 0; r < 8; ++r) {
    const int M = r + 8 * half;
    G[(u * K_ + mpair * 32 + M) * S_ + l16]      = a0p[r];
    G[(u * K_ + mpair * 32 + 16 + M) * S_ + l16] = a1p[r];
  }
}

// ---------------------------------------------------------------------------
// Kernel 6: b_ij[c,u] += (1/B) * sum_{s,i} W[c,u,s,i] * G[u][(c,i)][s]
// ---------------------------------------------------------------------------
__global__ void bij_update_kernel(const float* __restrict__ W,
                                  const float* __restrict__ G,
                                  float* __restrict__ bij) {
  int tid = blockIdx.x * blockDim.x + threadIdx.x;
  if (tid >= C_ * U_) return;
  int c = tid / U_;
  int u = tid % U_;
  float acc = 0.0f;
  for (int s = 0; s < S_; ++s) {
#pragma unroll
    for (int i = 0; i < IU; ++i) {
      acc += W[((c * U_ + u) * S_ + s) * IU + i] *
             G[(u * K_ + c * 8 + i) * S_ + s];
    }
  }
  bij[c * U_ + u] += acc * (1.0f / (float)B_);
}

// ---------------------------------------------------------------------------
extern "C" void kernel_launch(void* const* d_in, const int* in_sizes, int n_in,
                              void* d_out, int out_size, void* d_ws, size_t ws_size,
                              hipStream_t stream) {
  const float* x = (const float*)d_in[0];  // [B, IU, C]
  const float* W = (const float*)d_in[1];  // [C, U, S, IU]
  float* out = (float*)d_out;              // [B, U, S, 1] == [B,U,S] flat

  // Workspace layout (floats): Xt | bij | cij | P | v | G  (~18.2 MB total)
  float* Xt  = (float*)d_ws;                         // K_*B_       = 2,359,296
  float* bij = Xt  + (size_t)K_ * B_;                // C_*U_       =    11,520
  float* cij = bij + (size_t)C_ * U_;                // C_*U_       =    11,520
  float* P   = cij + (size_t)C_ * U_;                // KSPLIT*U*B*S=   655,360
  float* v   = P   + (size_t)KSPLIT * U_ * B_ * S_;  // B*U*S       =    40,960
  float* G   = v   + (size_t)B_ * U_ * S_;           // U*K_*S      = 1,474,560

  // Pack x -> Xt and zero b_ij
  pack_x_kernel<<<(K_ * B_ + 255) / 256, 256, 0, stream>>>(x, Xt, bij);

  const dim3 sj_grid(B_ / 32, KSPLIT, U_);     // 8 x 16 x 10 = 1280 waves, 2 acc each
  const dim3 g_grid(K_ / 32, U_);              // 288 x 10 = 2880 waves, 2 acc each

  for (int it = 0; it < 3; ++it) {
    softmax_u_kernel<<<(C_ + 127) / 128, 128, 0, stream>>>(bij, cij);
    sj_gemm_kernel<<<sj_grid, 32, 0, stream>>>(W, cij, Xt, P);
    squash_kernel<<<(B_ * U_ + 255) / 256, 256, 0, stream>>>(
        P, (it == 2) ? out : v);
    if (it < 2) {
      g_gemm_kernel<<<g_grid, 32, 0, stream>>>(Xt, v, G);
      bij_update_kernel<<<(C_ * U_ + 255) / 256, 256, 0, stream>>>(W, G, bij);
    }
  }
}